// DigitCapsLayer_14053132993112
// MI455X (gfx1250) — compile-verified
//
#include <hip/hip_runtime.h>

typedef float v2f __attribute__((ext_vector_type(2)));
typedef float v8f __attribute__((ext_vector_type(8)));

#define B_      256
#define P_      1152
#define J_      10
#define DIN     8
#define DOUT    16
#define THREADS 256
#define WAVES   8
#define PPW     (P_ / WAVES)      // 144 p's per wave for t-GEMM K reduction
#define TILES   (P_ / 16)         // 72 p-tiles
#define TPW     (TILES / WAVES)   // 9 tiles per wave for b-update

// LDS layout (float offsets); small pads so unmasked WMMA lane loads stay in-bounds
#define OFF_X    0                           // x[p][k]        9216 + 16 pad
#define OFF_W    (OFF_X + P_ * DIN + 16)     // W[j][d][k]     1280
#define OFF_B    (OFF_W + J_ * DOUT * DIN)   // b[p][j]        11520 + 16 pad
#define OFF_C    (OFF_B + P_ * J_ + 16)      // e[p][j]        11520 + 16 pad
#define OFF_RED  (OFF_C + P_ * J_ + 16)      // tree-reduce    2560
#define OFF_TP   (OFF_RED + THREADS * J_)    // t partials     8*256
#define OFF_T    (OFF_TP + WAVES * 256)      // t[j][k]        80
#define OFF_S    (OFF_T + J_ * DIN)          // s[j][d]        160
#define OFF_V    (OFF_S + J_ * DOUT)         // v[j][d]        160
#define OFF_G    (OFF_V + J_ * DOUT)         // g[j][k]        128 (padded, upper zero)
#define OFF_MX   (OFF_G + 128)               // max / squash scale  16
#define OFF_RS   (OFF_MX + 16)               // 1/sumexp       16
#define LDS_FLOATS (OFF_RS + 16)             // ~38752 floats = ~155 KB

__global__ __launch_bounds__(THREADS)
void digitcaps_routing_kernel(const float* __restrict__ xg,
                              const float* __restrict__ Wg,
                              float* __restrict__ out) {
  extern __shared__ float lds[];
  float* X  = lds + OFF_X;
  float* Wl = lds + OFF_W;
  float* Bb = lds + OFF_B;
  float* Cc = lds + OFF_C;
  float* RED = lds + OFF_RED;
  float* TP = lds + OFF_TP;
  float* TT = lds + OFF_T;
  float* SL = lds + OFF_S;
  float* VL = lds + OFF_V;
  float* GL = lds + OFF_G;
  float* MX = lds + OFF_MX;
  float* RS = lds + OFF_RS;

  const int tid  = threadIdx.x;
  const int lane = tid & 31;
  const int wv   = tid >> 5;
  const int jn   = lane & 15;   // WMMA: A row (M) or B/D column (N)
  const int hi   = lane >> 4;   // lane-half selects K pair / D row offset
  const int bIdx = blockIdx.x;

  // ---- stage x[b] (coalesced b128), W, zero b-logits ----
  const float4* xv = (const float4*)(xg + (size_t)bIdx * P_ * DIN);
  float4* x4 = (float4*)X;
  for (int i = tid; i < P_ * DIN / 4; i += THREADS) x4[i] = xv[i];
  for (int i = tid; i < J_ * DOUT * DIN; i += THREADS) Wl[i] = Wg[i];
  for (int i = tid; i < P_ * J_ + 16; i += THREADS) Bb[i] = 0.0f;
  for (int i = tid; i < 128; i += THREADS) GL[i] = 0.0f;  // pad cols j>=10 stay 0
  __syncthreads();

  for (int iter = 0; iter < 3; ++iter) {
    // ================= softmax over p (axis=1), per j =================
    float m[J_];
#pragma unroll
    for (int j = 0; j < J_; ++j) m[j] = -3.402823466e+38f;
    for (int p = tid; p < P_; p += THREADS) {
#pragma unroll
      for (int j = 0; j < J_; ++j) m[j] = fmaxf(m[j], Bb[p * J_ + j]);
    }
#pragma unroll
    for (int j = 0; j < J_; ++j) RED[tid * J_ + j] = m[j];
    __syncthreads();
    for (int off = 128; off > 0; off >>= 1) {
      if (tid < off) {
#pragma unroll
        for (int j = 0; j < J_; ++j)
          RED[tid * J_ + j] = fmaxf(RED[tid * J_ + j], RED[(tid + off) * J_ + j]);
      }
      __syncthreads();
    }
    if (tid < J_) MX[tid] = RED[tid];
    __syncthreads();

    float sm[J_];
#pragma unroll
    for (int j = 0; j < J_; ++j) sm[j] = 0.0f;
    for (int p = tid; p < P_; p += THREADS) {
#pragma unroll
      for (int j = 0; j < J_; ++j) {
        float e = __expf(Bb[p * J_ + j] - MX[j]);
        Cc[p * J_ + j] = e;   // unnormalized; 1/sum folded into t later
        sm[j] += e;
      }
    }
#pragma unroll
    for (int j = 0; j < J_; ++j) RED[tid * J_ + j] = sm[j];
    __syncthreads();
    for (int off = 128; off > 0; off >>= 1) {
      if (tid < off) {
#pragma unroll
        for (int j = 0; j < J_; ++j)
          RED[tid * J_ + j] += RED[(tid + off) * J_ + j];
      }
      __syncthreads();
    }
    if (tid < J_) RS[tid] = 1.0f / RED[tid];
    __syncthreads();

    // ====== t[j,k] = sum_p e[p,j] x[p,k] : WMMA f32 16x16x4, K=1152 ======
    {
      v8f acc = {};
      const int pbase = wv * PPW;
      for (int kk = 0; kk < PPW; kk += 4) {
        const int ph = pbase + kk + 2 * hi;   // K index for this lane half
        v2f a, bm;
        a.x  = Cc[ph * J_ + jn];              // A[M=j][K=p]
        a.y  = Cc[(ph + 1) * J_ + jn];
        bm.x = X[ph * DIN + jn];              // B[K=p][N=k]
        bm.y = X[(ph + 1) * DIN + jn];
        acc = __builtin_amdgcn_wmma_f32_16x16x4_f32(false, a, false, bm,
                                                    (short)0, acc, false, false);
      }
#pragma unroll
      for (int v = 0; v < 8; ++v)
        TP[wv * 256 + (v + 8 * hi) * 16 + jn] = acc[v];
    }
    __syncthreads();

    // combine per-wave partials, normalize by 1/sumexp
    if (tid < J_ * DIN) {
      const int j = tid >> 3, k = tid & 7;
      float tv = 0.0f;
#pragma unroll
      for (int w = 0; w < WAVES; ++w) tv += TP[w * 256 + j * 16 + k];
      TT[tid] = tv * RS[j];
    }
    __syncthreads();

    // s[j,d] = sum_k W[j,d,k] t[j,k]
    if (tid < J_ * DOUT) {
      const int j = tid >> 4;
      float s = 0.0f;
#pragma unroll
      for (int k = 0; k < DIN; ++k) s += Wl[tid * DIN + k] * TT[j * DIN + k];
      SL[tid] = s;
    }
    __syncthreads();

    // squash scale per j
    if (tid < J_) {
      float sq = 0.0f;
#pragma unroll
      for (int d = 0; d < DOUT; ++d) { float s = SL[tid * DOUT + d]; sq += s * s; }
      MX[tid] = sq / ((1.0f + sq) * sqrtf(sq + 1e-9f));  // reuse MX as scale
    }
    __syncthreads();

    if (tid < J_ * DOUT) {
      const int j = tid >> 4;
      const float v = SL[tid] * MX[j];
      VL[tid] = v;
      if (iter == 2) out[(size_t)bIdx * J_ * DOUT + tid] = v;
    }

    if (iter < 2) {
      __syncthreads();
      // g[j,k] = sum_d W[j,d,k] v[j,d]
      if (tid < J_ * DIN) {
        const int j = tid >> 3, k = tid & 7;
        float g = 0.0f;
#pragma unroll
        for (int d = 0; d < DOUT; ++d)
          g += Wl[(j * DOUT + d) * DIN + k] * VL[j * DOUT + d];
        GL[tid] = g;
      }
      __syncthreads();

      // ====== b[p,j] += sum_k x[p,k] g[j,k] : WMMA, 9 p-tiles per wave ======
      for (int tI = 0; tI < TPW; ++tI) {
        const int p0 = (wv * TPW + tI) * 16;
        v8f cacc;
#pragma unroll
        for (int v = 0; v < 8; ++v)
          cacc[v] = Bb[(p0 + v + 8 * hi) * J_ + jn];   // C = current b tile
#pragma unroll
        for (int k0 = 0; k0 < DIN; k0 += 4) {
          const int kh = k0 + 2 * hi;
          v2f a, bm;
          a.x  = X[(p0 + jn) * DIN + kh];              // A[M=p][K=k]
          a.y  = X[(p0 + jn) * DIN + kh + 1];
          bm.x = GL[jn * DIN + kh];                    // B[K=k][N=j] (pad rows = 0)
          bm.y = GL[jn * DIN + kh + 1];
          cacc = __builtin_amdgcn_wmma_f32_16x16x4_f32(false, a, false, bm,
                                                       (short)0, cacc, false, false);
        }
        if (jn < J_) {                                  // discard padded columns
#pragma unroll
          for (int v = 0; v < 8; ++v)
            Bb[(p0 + v + 8 * hi) * J_ + jn] = cacc[v];
        }
      }
      __syncthreads();
    }
  }
}

extern "C" void kernel_launch(void* const* d_in, const int* in_sizes, int n_in,
                              void* d_out, int out_size, void* d_ws, size_t ws_size,
                              hipStream_t stream) {
  const float* x = (const float*)d_in[0];   // [256,1152,8]
  const float* W = (const float*)d_in[1];   // [10,16,8]
  float* out = (float*)d_out;               // [256,10,16]
  (void)in_sizes; (void)n_in; (void)out_size; (void)d_ws; (void)ws_size;

  const size_t shmem = (size_t)LDS_FLOATS * sizeof(float);  // ~155 KB dynamic LDS
  hipFuncSetAttribute(reinterpret_cast<const void*>(digitcaps_routing_kernel),
                      hipFuncAttributeMaxDynamicSharedMemorySize, (int)shmem);
  digitcaps_routing_kernel<<<B_, THREADS, shmem, stream>>>(x, W, out);
}